// LinearSSM_90314572300304
// MI455X (gfx1250) — compile-verified
//
#include <hip/hip_runtime.h>
#include <hip/hip_bf16.h>
#include <stdint.h>

typedef __attribute__((ext_vector_type(2))) float v2f;
typedef __attribute__((ext_vector_type(8))) float v8f;

#define BATCH   64
#define T_STEPS 1024
#define SDIM    512     // state_dim
#define ADIM    128     // action_dim
#define XPAD    516     // 512 + 4  -> conflict-free ds_load_b64 fragment reads
#define UPAD    132     // 128 + 4

// One workgroup per 16-row batch strip (4 WGs; strips are independent through
// the recurrence). 1024 threads = 32 waves; wave n owns output columns
// [16n,16n+16). x_t [16 x 512] lives in LDS; u is double-buffered in LDS and
// prefetched one step ahead with GLOBAL_LOAD_ASYNC_TO_LDS_B64 (ASYNCcnt).
// A (1 MB) / B (256 KB) stream from the 192 MB L2 every step.
__global__ __launch_bounds__(1024, 1)
void ssm_rollout_f32wmma(const float* __restrict__ x0,
                         const float* __restrict__ us,
                         const float* __restrict__ A,
                         const float* __restrict__ B,
                         float* __restrict__ out)
{
    __shared__ float lds_x[16 * XPAD];        // ~33.0 KB
    __shared__ float lds_u[2][16 * UPAD];     // ~16.9 KB double buffer

    const int tid  = threadIdx.x;
    const int lane = tid & 31;
    const int wid  = tid >> 5;           // 0..31 : N-tile index
    const int m    = blockIdx.x;         // 0..3  : batch strip
    const int n0   = wid << 4;           // first output column of this wave
    const int half = lane >> 4;          // WMMA frag half: K pair select
    const int l    = lane & 15;          // row (A-frag) / column (B-frag) id

    // Per-thread async-staging coordinates: thread covers u[row r][2c, 2c+1].
    const int ur = tid >> 6;             // 0..15
    const int uc = (tid & 63) << 1;      // 0,2,..,126
    const float* u_gbase =
        us + ((size_t)(m * 16 + ur) * T_STEPS) * ADIM + uc;   // + t*ADIM later
    const uint32_t u_lds_off0 =
        (uint32_t)(uintptr_t)(&lds_u[0][ur * UPAD + uc]);     // LDS byte offset
    const uint32_t u_lds_stride = (uint32_t)(16 * UPAD * sizeof(float));

    // ---- preload x0 strip [16 x 512] into LDS (coalesced) ----
    for (int e = tid; e < 16 * SDIM; e += 1024) {
        int row = e >> 9, col = e & (SDIM - 1);
        lds_x[row * XPAD + col] = x0[(size_t)(m * 16 + row) * SDIM + col];
    }
    // ---- async-stage u_0 into buffer 0 ----
    {
        const float* g = u_gbase;        // t = 0
        asm volatile("global_load_async_to_lds_b64 %0, %1, off"
                     :: "v"(u_lds_off0), "v"(g) : "memory");
        asm volatile("s_wait_asynccnt 0x0" ::: "memory");
    }
    __syncthreads();

    for (int t = 0; t < T_STEPS; ++t) {
        const int buf = t & 1;

        // ---- prefetch u_{t+1} into the other buffer (overlaps compute) ----
        if (t + 1 < T_STEPS) {
            const float* g = u_gbase + (size_t)(t + 1) * ADIM;
            const uint32_t loff = u_lds_off0 + (buf ^ 1) * u_lds_stride;
            asm volatile("global_load_async_to_lds_b64 %0, %1, off"
                         :: "v"(loff), "v"(g) : "memory");
        }

        const float* ub = &lds_u[buf][0];
        v8f acc = {};   // 16x16 f32 accumulator tile (8 VGPRs)

        // ---- acc += u_t @ B   (K = 128, f32 WMMA 16x16x4) ----
        {
            const float* Bp = B + n0 + l;
            #pragma unroll 4
            for (int k0 = 0; k0 < ADIM; k0 += 4) {
                const int k = k0 + 2 * half;
                v2f a, b;
                a.x = ub[l * UPAD + k];             // u[row=l][k]
                a.y = ub[l * UPAD + k + 1];
                b.x = Bp[(size_t)k * SDIM];         // B[k][n0+l]
                b.y = Bp[(size_t)(k + 1) * SDIM];
                acc = __builtin_amdgcn_wmma_f32_16x16x4_f32(
                          false, a, false, b, (short)0, acc, false, false);
            }
        }

        // ---- acc += x_t @ A   (K = 512, f32 WMMA 16x16x4) ----
        {
            const float* Ap = A + n0 + l;
            #pragma unroll 4
            for (int k0 = 0; k0 < SDIM; k0 += 4) {
                const int k = k0 + 2 * half;
                v2f a, b;
                a.x = lds_x[l * XPAD + k];          // x[row=l][k]
                a.y = lds_x[l * XPAD + k + 1];
                b.x = Ap[(size_t)k * SDIM];         // A[k][n0+l]
                b.y = Ap[(size_t)(k + 1) * SDIM];
                acc = __builtin_amdgcn_wmma_f32_16x16x4_f32(
                          false, a, false, b, (short)0, acc, false, false);
            }
        }

        __syncthreads();   // all waves finished reading x_t

        // ---- commit x_{t+1}: back into LDS + to global output ----
        #pragma unroll
        for (int v = 0; v < 8; ++v) {
            const int row = v + 8 * half;           // C-tile row for this VGPR
            const float val = acc[v];
            lds_x[row * XPAD + n0 + l] = val;
            out[((size_t)(m * 16 + row) * T_STEPS + t) * SDIM + n0 + l] = val;
        }

        // our async u_{t+1} writes must be in LDS before anyone reads them
        asm volatile("s_wait_asynccnt 0x0" ::: "memory");
        __syncthreads();   // x_{t+1} + u_{t+1} visible to all waves
    }
}

extern "C" void kernel_launch(void* const* d_in, const int* in_sizes, int n_in,
                              void* d_out, int out_size, void* d_ws, size_t ws_size,
                              hipStream_t stream) {
    (void)in_sizes; (void)n_in; (void)d_ws; (void)ws_size; (void)out_size;
    const float* x0 = (const float*)d_in[0];   // [64, 512]
    const float* us = (const float*)d_in[1];   // [64, 1024, 128]
    const float* A  = (const float*)d_in[2];   // [512, 512]
    const float* B  = (const float*)d_in[3];   // [128, 512]
    float* out = (float*)d_out;                // [64, 1024, 512]

    ssm_rollout_f32wmma<<<dim3(BATCH / 16), dim3(1024), 0, stream>>>(
        x0, us, A, B, out);
}